// KendallDependenceMetric_4552665334328
// MI455X (gfx1250) — compile-verified
//
#include <hip/hip_runtime.h>

#define N   8192
#define UT  256   // u elements per block (= threads per block)
#define VT  256   // v elements per block (LDS slab)

typedef unsigned int v4u __attribute__((ext_vector_type(4)));
typedef int          v8i __attribute__((ext_vector_type(8)));
typedef int          v4i __attribute__((ext_vector_type(4)));

// ---------------------------------------------------------------------------
// CDNA5 Tensor Data Mover: async 1-D copy of `nelem` f32 from global -> LDS.
// D# layout per CDNA5 ISA (2-D tensor, groups 2/3
// unused). Tracked by TENSORcnt. (Descriptor lowering verified in round-1 asm.)
// ---------------------------------------------------------------------------
__device__ __forceinline__ void tdm_load_f32_1d(unsigned int lds_off,
                                                const void* gptr,
                                                unsigned int nelem)
{
    unsigned long long ga = (unsigned long long)(uintptr_t)gptr;

    v4u g0;
    g0[0] = 1u;                                   // count=1 (valid descriptor)
    g0[1] = lds_off;                              // lds_addr (bytes)
    g0[2] = (unsigned int)ga;                     // global_addr[31:0]
    g0[3] = ((unsigned int)(ga >> 32) & 0x01FFFFFFu) | (2u << 30); // [56:32] | type=2

    v8i g1;
    g1[0] = 0x00020000;                 // data_size=2 (4 bytes); no mask/pad/iterate
    g1[1] = (int)(nelem << 16);         // tensor_dim0[15:0] at bits [31:16]
    g1[2] = 0x00010000;                 // tensor_dim0[31:16]=0, tensor_dim1=1
    g1[3] = (int)(nelem << 16);         // tensor_dim1[31:16]=0, tile_dim0=nelem
    g1[4] = 0x00000001;                 // tile_dim1=1, tile_dim2=0
    g1[5] = (int)nelem;                 // tensor_dim0_stride[31:0]
    g1[6] = 0;                          // stride0[47:32]=0, stride1[15:0]=0
    g1[7] = 0;                          // stride1[47:16]=0

    v4i gz = {0, 0, 0, 0};
#if defined(__clang_major__) && (__clang_major__ >= 23)
    v8i gz8 = {0, 0, 0, 0, 0, 0, 0, 0};
    __builtin_amdgcn_tensor_load_to_lds(g0, g1, gz, gz, gz8, 0);
#else
    __builtin_amdgcn_tensor_load_to_lds(g0, g1, gz, gz, 0);
#endif
}

// 32 pairs in ~5 ops: two wave32 compare-masks, xor, popcount, accumulate.
__device__ __forceinline__ void count32(float xu, float yu, float xv, float yv,
                                        unsigned int& local)
{
    const unsigned int m1 = (unsigned int)__ballot(xu > xv);
    const unsigned int m2 = (unsigned int)__ballot(yu > yv);
    local += (unsigned int)__popc(m1 ^ m2);
}

// ---------------------------------------------------------------------------
// Pair kernel: block (bx,by) counts discordant ordered pairs for
// u in [256*bx, +256) x v in [256*by, +256). Diagonal contributes 0 naturally.
// ---------------------------------------------------------------------------
__global__ void __launch_bounds__(UT)
kendall_pairs(const float* __restrict__ X, const float* __restrict__ Y,
              unsigned int* __restrict__ cnt)
{
    __shared__ alignas(16) float sX[VT];
    __shared__ alignas(16) float sY[VT];

    const int u  = blockIdx.x * UT + (int)threadIdx.x;
    const int v0 = blockIdx.y * VT;

    // Async-stage the v-slabs into LDS with the TDM (each wave issues its own
    // idempotent copy and waits on its own TENSORcnt; identical bytes).
    tdm_load_f32_1d((unsigned int)(uintptr_t)(void*)sX, X + v0, VT);
    tdm_load_f32_1d((unsigned int)(uintptr_t)(void*)sY, Y + v0, VT);

    // Overlap: fetch this lane's u-values while the DMA runs.
    const float xu = X[u];
    const float yu = Y[u];

    __builtin_amdgcn_s_wait_tensorcnt(0);
    __syncthreads();

    unsigned int local = 0;
    // Manual unroll x4: one ds_load_b128 per slab per 128 pairs.
    for (int k = 0; k < VT; k += 4) {
        const float4 xv4 = *(const float4*)&sX[k];   // LDS broadcast, B128
        const float4 yv4 = *(const float4*)&sY[k];
        count32(xu, yu, xv4.x, yv4.x, local);
        count32(xu, yu, xv4.y, yv4.y, local);
        count32(xu, yu, xv4.z, yv4.z, local);
        count32(xu, yu, xv4.w, yv4.w, local);
    }

    // Every lane in a wave holds the same total -> one atomic per wave.
    if ((threadIdx.x & 31u) == 0u)
        atomicAdd(cnt, local);
}

__global__ void zero_counter(unsigned int* __restrict__ c)
{
    *c = 0u;
}

__global__ void finalize(const unsigned int* __restrict__ c,
                         float* __restrict__ out)
{
    out[0] = (float)(*c) / (float)((long long)N * (long long)(N - 1));
}

extern "C" void kernel_launch(void* const* d_in, const int* in_sizes, int n_in,
                              void* d_out, int out_size, void* d_ws, size_t ws_size,
                              hipStream_t stream)
{
    const float* X = (const float*)d_in[0];      // X
    const float* Y = (const float*)d_in[1];      // X_hat
    unsigned int* cnt = (unsigned int*)d_ws;     // 4-byte counter in scratch
    float* out = (float*)d_out;

    zero_counter<<<dim3(1), dim3(1), 0, stream>>>(cnt);

    dim3 grid(N / UT, N / VT);                   // 32 x 32 blocks
    kendall_pairs<<<grid, dim3(UT), 0, stream>>>(X, Y, cnt);

    finalize<<<dim3(1), dim3(1), 0, stream>>>(cnt, out);
}